// CustomISTFT_12171937317492
// MI455X (gfx1250) — compile-verified
//
#include <hip/hip_runtime.h>
#include <math.h>

// ---------------------------------------------------------------------------
// ISTFT for MI455X (gfx1250, wave32, WMMA).
//
//  Re fr[m,t] = (1/N) sum_k w_k ( a[k,t] cos(2*pi*k*m/N) - b[k,t] sin(...) )
//               -> fp32 GEMM via V_WMMA_F32_16X16X4_F32
//  Im fr[m,t] = (1/N) ( b[0,t] + (-1)^m b[N/2,t] )   -> computed analytically
//  then windowed overlap-add (hop = N/4), window-sum normalize, center crop.
//
// LDS-staged B operand: each block stages a 32(k) x 128(t) chunk of a/b into
// LDS once per iteration via GLOBAL_LOAD_ASYNC_TO_LDS_B128 (ASYNCcnt), cutting
// the 4x redundant L2 reads of the register-direct version. LDS row stride 144
// dwords -> (k,k+1) pair is 144 dwords apart (ds_load_2addr-fusable) and the
// 16-bank/row skew makes the two lane-halves conflict-free.
// ---------------------------------------------------------------------------

typedef __attribute__((ext_vector_type(2))) float v2f;   // A/B operand: 16x4 / 4x16 f32
typedef __attribute__((ext_vector_type(8))) float v8f;   // C/D operand: 16x16 f32
typedef int v4i __attribute__((vector_size(16)));        // async-copy payload type

#define NFFT   4096
#define FBINS  2049
#define TFR    4096
#define HOP    1024
#define OUTLEN 4193280LL   // (TFR-1)*HOP + NFFT - NFFT  (center crop)

#define CHUNK_K   32       // k-bins staged per block iteration
#define NCHUNKS   65       // 65*32 = 2080 >= 2049 (zero weights pad the tail)
#define LDS_RS    144      // padded LDS row stride (dwords)

#if defined(__has_builtin)
#  if __has_builtin(__builtin_amdgcn_global_load_async_to_lds_b128)
#    define USE_ASYNC_LDS 1
#  endif
#  if __has_builtin(__builtin_amdgcn_s_wait_asynccnt)
#    define HAVE_WAIT_ASYNC 1
#  endif
#endif

__device__ __forceinline__ void stage16B(const float* __restrict__ g, float* l)
{
#ifdef USE_ASYNC_LDS
    // (global src, LDS dst, imm offset, cpol) -- payload typed as 4xi32.
    __builtin_amdgcn_global_load_async_to_lds_b128((v4i*)g, (v4i*)l, 0, 0);
#else
    *(float4*)l = *(const float4*)g;
#endif
}

__device__ __forceinline__ void wait_stage()
{
#ifdef USE_ASYNC_LDS
#  ifdef HAVE_WAIT_ASYNC
    __builtin_amdgcn_s_wait_asynccnt(0);
#  else
    asm volatile("s_wait_asynccnt 0" ::: "memory");
#  endif
#endif
}

// ---------------------------------------------------------------------------
// Kernel 1: R[m, t] = Re(ifft(full_spectrum))[m] for frame t, as WMMA GEMM.
// Block: 256 threads = 8 wave32 arranged 4(m) x 2(t-group).
// Block tile: 64 rows (m) x 128 cols (t). Each wave: 16x64 strip, 4 accums
// sharing one basis (A) pair -> 2 sincos feed 8 WMMAs per K-step.
// Basis cos/sin generated on the fly from exact integer phase (m*k) & 4095.
// EXEC stays all ones everywhere (WMMA requirement): clamps, no branches.
// ---------------------------------------------------------------------------
__global__ __launch_bounds__(256) void istft_idft_wmma(const float* __restrict__ z,
                                                       float* __restrict__ R)
{
    __shared__ float lds_a[CHUNK_K * LDS_RS];   // 18 KB
    __shared__ float lds_b[CHUNK_K * LDS_RS];   // 18 KB

    const int tid    = threadIdx.x;
    const int wave   = tid >> 5;           // 0..7
    const int lane   = tid & 31;
    const int laneLo = lane & 15;
    const int hi     = lane >> 4;          // lane half selects K pair (A/B layout)

    const int waveM = wave & 3;            // 4 m-subtiles per block
    const int waveT = wave >> 2;           // 2 t-groups per block

    const int mTile  = (blockIdx.x << 6) + (waveM << 4);
    const int tBlock = (blockIdx.y << 7);                 // 128-wide block tile
    const int cL     = (waveT << 6) + laneLo;             // LDS column, 0..127

    const float* __restrict__ a = z;                          // z[0] : real
    const float* __restrict__ b = z + (size_t)FBINS * TFR;    // z[1] : imag

    const int m = mTile + laneLo;   // A-matrix row held by this lane

    // Staging assignment: 32 col-groups (16B) x 8 row-slots; 4 rows per thread.
    const int sc = (tid & 31) << 2;   // col group 0,4,...,124
    const int sk = tid >> 5;          // row slot 0..7 (rows sk, sk+8, sk+16, sk+24)

    v8f acc0 = {}, acc1 = {}, acc2 = {}, acc3 = {};
    const float twopi_n = 6.28318530717958647692f / (float)NFFT;

    for (int chunk = 0; chunk < NCHUNKS; ++chunk) {
        const int k0c = chunk * CHUNK_K;

        __syncthreads();   // previous chunk's LDS reads complete
#pragma unroll
        for (int i = 0; i < 4; ++i) {
            const int kloc = sk + (i << 3);
            const int g    = min(k0c + kloc, 2048);        // clamp padded tail
            const size_t go = (size_t)g * TFR + tBlock + sc;
            float* la = &lds_a[kloc * LDS_RS + sc];
            float* lb = &lds_b[kloc * LDS_RS + sc];
            stage16B(a + go, la);
            stage16B(b + go, lb);
        }
        wait_stage();
        __syncthreads();   // whole chunk staged

        // Prefetch next chunk's rows while we compute (global_prefetch_b8).
        {
            const int gn = min(k0c + CHUNK_K + sk, 2048);
            __builtin_prefetch(a + (size_t)gn * TFR + tBlock + sc, 0, 0);
            __builtin_prefetch(b + (size_t)gn * TFR + tBlock + sc, 0, 0);
        }

#pragma unroll
        for (int s = 0; s < CHUNK_K / 4; ++s) {
            const int kAl = (s << 2) + (hi << 1);   // local k: lanes0-15 K0/K1, 16-31 K2/K3
            const int kA  = k0c + kAl;
            const int kB  = kA + 1;

            // Hermitian weights folded with 1/N; zero weight kills the pad tail.
            float wA = (kA == 0 || kA == 2048) ? 1.0f : (kA > 2048 ? 0.0f : 2.0f);
            float wB = (kB == 2048)            ? 1.0f : (kB > 2048 ? 0.0f : 2.0f);
            wA *= (1.0f / (float)NFFT);
            wB *= (1.0f / (float)NFFT);

            // Exact integer phase keeps the sincos argument in [0, 2*pi).
            const int pA = (m * kA) & (NFFT - 1);
            const int pB = (m * kB) & (NFFT - 1);
            float sA, cA, sB, cB;
            __sincosf((float)pA * twopi_n, &sA, &cA);
            __sincosf((float)pB * twopi_n, &sB, &cB);

            v2f Ac, As;
            Ac.x =  wA * cA;  Ac.y =  wB * cB;    // cos basis tile (A, 16x4)
            As.x = -wA * sA;  As.y = -wB * sB;    // -sin basis tile

            const float* __restrict__ ra = &lds_a[kAl * LDS_RS + cL];
            const float* __restrict__ rb = &lds_b[kAl * LDS_RS + cL];

            v2f Ba0, Ba1, Ba2, Ba3, Bb0, Bb1, Bb2, Bb3;
            Ba0.x = ra[ 0];  Ba0.y = ra[LDS_RS     ];
            Ba1.x = ra[16];  Ba1.y = ra[LDS_RS + 16];
            Ba2.x = ra[32];  Ba2.y = ra[LDS_RS + 32];
            Ba3.x = ra[48];  Ba3.y = ra[LDS_RS + 48];
            Bb0.x = rb[ 0];  Bb0.y = rb[LDS_RS     ];
            Bb1.x = rb[16];  Bb1.y = rb[LDS_RS + 16];
            Bb2.x = rb[32];  Bb2.y = rb[LDS_RS + 32];
            Bb3.x = rb[48];  Bb3.y = rb[LDS_RS + 48];

            // One basis pair feeds 8 WMMAs (4 columns x {cos,sin}).
            acc0 = __builtin_amdgcn_wmma_f32_16x16x4_f32(false, Ac, false, Ba0,
                                                         (short)0, acc0, false, false);
            acc1 = __builtin_amdgcn_wmma_f32_16x16x4_f32(false, Ac, false, Ba1,
                                                         (short)0, acc1, false, false);
            acc2 = __builtin_amdgcn_wmma_f32_16x16x4_f32(false, Ac, false, Ba2,
                                                         (short)0, acc2, false, false);
            acc3 = __builtin_amdgcn_wmma_f32_16x16x4_f32(false, Ac, false, Ba3,
                                                         (short)0, acc3, false, false);
            acc0 = __builtin_amdgcn_wmma_f32_16x16x4_f32(false, As, false, Bb0,
                                                         (short)0, acc0, false, false);
            acc1 = __builtin_amdgcn_wmma_f32_16x16x4_f32(false, As, false, Bb1,
                                                         (short)0, acc1, false, false);
            acc2 = __builtin_amdgcn_wmma_f32_16x16x4_f32(false, As, false, Bb2,
                                                         (short)0, acc2, false, false);
            acc3 = __builtin_amdgcn_wmma_f32_16x16x4_f32(false, As, false, Bb3,
                                                         (short)0, acc3, false, false);
        }
    }

    // C/D layout: VGPR r -> row (r + 8*laneHalf), col = lane&15.
    const int row0 = mTile + (hi << 3);
#pragma unroll
    for (int r = 0; r < 8; ++r) {
        float* __restrict__ dst = R + (size_t)(row0 + r) * TFR
                                    + tBlock + (waveT << 6) + laneLo;
        dst[ 0] = acc0[r];
        dst[16] = acc1[r];
        dst[32] = acc2[r];
        dst[48] = acc3[r];
    }
}

// ---------------------------------------------------------------------------
// Kernel 2: gather-form overlap-add + window-sum normalize + center crop.
// Each cropped output sample o (p = o + NFFT/2) receives <=4 frame
// contributions: t = p/HOP - j, m = p - HOP*t, j = 0..3.
// Channel 0 reads R; channel 1 uses the closed-form imaginary part.
// ---------------------------------------------------------------------------
__global__ __launch_bounds__(256) void istft_ola(const float* __restrict__ z,
                                                 const float* __restrict__ window,
                                                 const float* __restrict__ R,
                                                 float* __restrict__ out)
{
    const long long o = (long long)blockIdx.x * blockDim.x + threadIdx.x;
    if (o >= OUTLEN) return;
    const long long p = o + (NFFT / 2);
    const int tmax = (int)(p >> 10);   // p / HOP

    const float* __restrict__ b0 = z + (size_t)FBINS * TFR;          // z[1][0][:]
    const float* __restrict__ bN = b0 + (size_t)2048 * TFR;          // z[1][2048][:]

    float s0 = 0.0f, s1 = 0.0f, ws = 0.0f;
#pragma unroll
    for (int j = 0; j < 4; ++j) {
        const int t = tmax - j;
        if (t < 0 || t >= TFR) continue;
        const int m = (int)(p - ((long long)t << 10));               // 0..4095
        const float w = window[m];
        ws += w;
        s0 += w * R[(size_t)m * TFR + t];
        const float im = (b0[t] + ((m & 1) ? -bN[t] : bN[t])) * (1.0f / (float)NFFT);
        s1 += w * im;
    }
    if (ws >= 1e-6f) { s0 /= ws; s1 /= ws; }
    out[o]          = s0;
    out[OUTLEN + o] = s1;
}

// ---------------------------------------------------------------------------
extern "C" void kernel_launch(void* const* d_in, const int* in_sizes, int n_in,
                              void* d_out, int out_size, void* d_ws, size_t ws_size,
                              hipStream_t stream)
{
    (void)in_sizes; (void)n_in; (void)out_size; (void)ws_size;
    const float* z      = (const float*)d_in[0];   // [2, 2049, 4096] f32
    const float* window = (const float*)d_in[1];   // [4096] f32
    float*       R      = (float*)d_ws;            // [4096, 4096] f32 scratch (64 MiB)
    float*       out    = (float*)d_out;           // [2, 4193280] f32

    dim3 g1(NFFT / 64, TFR / 128);  // 64 x 32 blocks, 8 waves each
    dim3 b1(256);
    istft_idft_wmma<<<g1, b1, 0, stream>>>(z, R);

    const int threads = 256;
    const int blocks  = (int)((OUTLEN + threads - 1) / threads);
    istft_ola<<<blocks, threads, 0, stream>>>(z, window, R, out);
}